// kernel_generated_2_62904091017236
// MI455X (gfx1250) — compile-verified
//
#include <hip/hip_runtime.h>

// ---------------------------------------------------------------------------
// Problem constants (from reference):
//   x:      (128, 64, 8, 14, 14) f32
//   w_conv: (8, 3, 3, 32)        f32   -> flat [kappa=g*9+k2*3+k1][j]
//   w_mix:  (64, 16)             f32
//   out:    (128, 512, 14, 14)   f32,  c = j*16 + p, rolled by +1 along h1
// Stage 1: t4[b,g,h1,h2,p] = sum_j x[b,j,g,h1,h2] * w_mix[j,p]
//          GEMM  M=128*8*196=200704 (p fastest in output), N=16, K=64
// Stage 2: t6[b,h1,h2,p,j] = sum_kappa maskedT4 * w_conv[kappa][j]
//          GEMM  M=(b,h1,h2,p), N=32, K=72
// ---------------------------------------------------------------------------

typedef __attribute__((ext_vector_type(2))) float v2f;
typedef __attribute__((ext_vector_type(8))) float v8f;

__device__ __forceinline__ v8f wmma_f32_4(v2f a, v2f b, v8f c) {
  return __builtin_amdgcn_wmma_f32_16x16x4_f32(false, a, false, b, (short)0, c,
                                               false, false);
}

// ------------------------------ Stage 1 ------------------------------------
// One wave per 16-row M-tile. 12544 tiles = 1568 blocks * 8 waves.
__global__ __launch_bounds__(256) void stage1_mix(const float* __restrict__ x,
                                                  const float* __restrict__ wmix,
                                                  float* __restrict__ t4) {
  const int tid  = threadIdx.x;
  const int lane = tid & 31;
  const int wave = tid >> 5;
  const int tile = blockIdx.x * 8 + wave;      // [0, 12544)

  const int half = (lane >> 4) & 1;            // 0: lanes 0-15, 1: lanes 16-31
  const int n    = lane & 15;                  // N / M-within-tile index
  const int sel  = half * 2;                   // K offset for this lane half

  // B operand: w_mix [64][16] held in registers (16 k-steps, v2f each).
  v2f bv[16];
#pragma unroll
  for (int kk = 0; kk < 16; ++kk) {
    const int j = kk * 4 + sel;
    bv[kk].x = wmix[j * 16 + n];
    bv[kk].y = wmix[(j + 1) * 16 + n];
  }

  // A row for this lane: row = tile*16 + (lane & 15), both halves hold M=0..15.
  const int row = tile * 16 + n;
  const int b   = row / 1568;                  // 8*196
  const int rem = row - b * 1568;
  const int g   = rem / 196;
  const int hh  = rem - g * 196;
  const int xoff = b * (64 * 1568) + g * 196 + hh;   // + j*1568

  v8f acc = {0.f, 0.f, 0.f, 0.f, 0.f, 0.f, 0.f, 0.f};
#pragma unroll
  for (int kk = 0; kk < 16; ++kk) {
    const int j = kk * 4 + sel;
    v2f a;
    a.x = x[xoff + j * 1568];
    a.y = x[xoff + (j + 1) * 1568];
    acc = wmma_f32_4(a, bv[kk], acc);
  }

  // D: VGPR v -> M = v + half*8, N = n.  t4 flat [row][16].
  const int rowD = tile * 16 + half * 8;
#pragma unroll
  for (int v = 0; v < 8; ++v) {
    t4[(rowD + v) * 16 + n] = acc[v];
  }
}

// ------------------------------ Stage 2 ------------------------------------
// One workgroup per (b, h1): 7 waves, each wave does 2 h2-tiles (14 total).
// Per tile: D = 16(p) x 32(j), K = 72 -> 18 k-steps x 2 N-tiles of WMMA.
__global__ __launch_bounds__(224) void stage2_conv(const float* __restrict__ t4,
                                                   const float* __restrict__ wconv,
                                                   float* __restrict__ out) {
  __shared__ float ldsW[72 * 32];     //  9 KB: w_conv as [kappa][j]
  __shared__ float ldsT[512 * 14];    // 28 KB: t6 tile as [c][h2]

  const int tid = threadIdx.x;
  const int b   = blockIdx.x / 14;
  const int h1  = blockIdx.x - b * 14;
  const int h1out = (h1 + 1) % 14;    // jnp.roll(+1) folded into the store

  for (int i = tid; i < 72 * 32; i += 224) ldsW[i] = wconv[i];
  __syncthreads();

  const int wave = tid >> 5;
  const int lane = tid & 31;
  const int half = (lane >> 4) & 1;
  const int n    = lane & 15;          // p for A, j-within-Ntile for B/D
  const int sel  = half * 2;

  // t4 flat index: b*25088 + g*3136 + h1*224 + d*16 + p
  const int tb = b * 25088 + h1 * 224 + n;

  // Masked A element. kLo is compile-time after unrolling; the lane-half
  // select turns the kappa decode into cndmasks between literals.
  auto aelem = [&](int kLo, int h2) -> float {
    const int kHi = kLo + 2;
    const int g  = half ? (kHi / 9) : (kLo / 9);
    const int k2 = half ? ((kHi / 3) % 3) : ((kLo / 3) % 3);
    const int k1 = half ? (kHi % 3) : (kLo % 3);
    const int mid = h2 + k2 - 1;            // first-unfold pad check
    const int d   = mid + k1 - 1;           // combined source index
    float v = 0.f;
    if (mid >= 0 && mid < 14 && d >= 0 && d < 14)
      v = t4[tb + g * 3136 + d * 16];
    return v;
  };

  for (int rep = 0; rep < 2; ++rep) {
    const int h2 = wave * 2 + rep;          // [0, 14)
    v8f acc0 = {0.f, 0.f, 0.f, 0.f, 0.f, 0.f, 0.f, 0.f};
    v8f acc1 = {0.f, 0.f, 0.f, 0.f, 0.f, 0.f, 0.f, 0.f};

#pragma unroll
    for (int kk = 0; kk < 18; ++kk) {
      v2f a;
      a.x = aelem(kk * 4 + 0, h2);
      a.y = aelem(kk * 4 + 1, h2);

      const int kA = kk * 4 + sel;          // B rows for this lane half
      v2f b0, b1;
      b0.x = ldsW[kA * 32 + n];
      b0.y = ldsW[(kA + 1) * 32 + n];
      b1.x = ldsW[kA * 32 + 16 + n];
      b1.y = ldsW[(kA + 1) * 32 + 16 + n];

      acc0 = wmma_f32_4(a, b0, acc0);       // N-tile j = 0..15
      acc1 = wmma_f32_4(a, b1, acc1);       // N-tile j = 16..31
    }

    // Scatter D into the LDS transpose buffer: row (p + 16*Ntile)*16 + j%16, pitch 14 (h2).
#pragma unroll
    for (int v = 0; v < 8; ++v) {
      const int j0 = v + half * 8;
      ldsT[((j0)      * 16 + n) * 14 + h2] = acc0[v];
      ldsT[((j0 + 16) * 16 + n) * 14 + h2] = acc1[v];
    }
  }

  __syncthreads();

  // Block-wide copy out: out[b, c, h1out, h2], h2 contiguous runs of 14.
  const int ob = b * 100352 + h1out * 14;   // 512*196 per batch
  for (int i = tid; i < 512 * 14; i += 224) {
    const int c  = i / 14;
    const int h2 = i - c * 14;
    out[ob + c * 196 + h2] = ldsT[i];
  }
}

// ---------------------------------------------------------------------------
extern "C" void kernel_launch(void* const* d_in, const int* in_sizes, int n_in,
                              void* d_out, int out_size, void* d_ws, size_t ws_size,
                              hipStream_t stream) {
  const float* x      = (const float*)d_in[0];   // (128,64,8,14,14)
  const float* w_conv = (const float*)d_in[1];   // (8,3,3,32)
  const float* w_mix  = (const float*)d_in[2];   // (64,16)
  float* out = (float*)d_out;                    // (128,512,14,14)
  float* t4  = (float*)d_ws;                     // 3,211,264 floats (12.25 MB)

  stage1_mix<<<1568, 256, 0, stream>>>(x, w_mix, t4);
  stage2_conv<<<1792, 224, 0, stream>>>(t4, w_conv, out);
}